// RGCN_26826365731108
// MI455X (gfx1250) — compile-verified
//
#include <hip/hip_runtime.h>

typedef __attribute__((ext_vector_type(2))) float v2f;
typedef __attribute__((ext_vector_type(4))) float v4f;
typedef __attribute__((ext_vector_type(8))) float v8f;

#define N_NODES 50000
#define E_EDGES 800000
#define D 100
#define B_BASES 10
#define LDH 1120           // 70 column tiles of 16: cols 0..999 = bases, 1000..1099 = root, 1100..1119 = pad
#define NTILE_GROUPS 14    // 70 tiles / 5 tiles-per-wave
#define MTILES 3125        // 50000 / 16 exactly
#define NXD (N_NODES * D)  // 5,000,000

// ---------------- utility kernels ----------------

__global__ void zero_f32(float* __restrict__ p, int n) {
    int i = blockIdx.x * blockDim.x + threadIdx.x;
    if (i < n) p[i] = 0.0f;
}

__global__ void gather_emb(const float* __restrict__ emb, const int* __restrict__ entity,
                           float* __restrict__ x) {
    int i = blockIdx.x * blockDim.x + threadIdx.x;
    if (i >= NXD) return;
    int n = i / D, d = i - n * D;
    x[i] = emb[(size_t)entity[n] * D + d];
}

// Pack Wp[k, j] (100 x 1120): j<1000 -> basis[b=j/100][k][d=j%100]; 1000<=j<1100 -> root[k][j-1000]; else 0
__global__ void pack_weights(const float* __restrict__ basis, const float* __restrict__ root,
                             float* __restrict__ Wp) {
    int i = blockIdx.x * blockDim.x + threadIdx.x;
    if (i >= D * LDH) return;
    int k = i / LDH, j = i - k * LDH;
    float v = 0.0f;
    if (j < 1000) {
        int b = j / 100, d = j - b * 100;
        v = basis[(size_t)b * D * D + (size_t)k * D + d];
    } else if (j < 1100) {
        v = root[(size_t)k * D + (j - 1000)];
    }
    Wp[i] = v;
}

__global__ void degree_hist(const int* __restrict__ dst, float* __restrict__ deg) {
    int e = blockIdx.x * blockDim.x + threadIdx.x;
    if (e < E_EDGES) atomicAdd(&deg[dst[e]], 1.0f);
}

// ---------------- f32 WMMA GEMM: Hext[N,1120] = X[N,100] @ Wp[100,1120] ----------------
// One wave computes a 16x80 output strip (5 tiles of 16x16), K-loop in steps of 4.
// A-frag (16x4 f32): lane 0-15 holds A[lane, k0..k0+1], lane 16-31 holds A[lane-16, k0+2..k0+3].
// B-frag (4x16 f32): lane 0-15 holds B[k0..k0+1, lane], lane 16-31 holds B[k0+2..k0+3, lane-16].
// D (16x16 f32, 8 VGPRs): vgpr v, lanes 0-15 -> row v, lanes 16-31 -> row v+8, col = lane&15.
__global__ __launch_bounds__(256) void gemm_wmma_f32(const float* __restrict__ X,
                                                     const float* __restrict__ W,
                                                     float* __restrict__ H) {
    int wid  = (blockIdx.x * blockDim.x + threadIdx.x) >> 5;
    if (wid >= MTILES * NTILE_GROUPS) return;
    int lane = threadIdx.x & 31;
    int l16  = lane & 15;
    int half = lane >> 4;

    int mt = wid / NTILE_GROUPS;
    int ng = wid - mt * NTILE_GROUPS;
    int m0 = mt * 16;
    int n0 = ng * 80;

    const float* xrow = X + (size_t)(m0 + l16) * D;

    v8f acc[5];
#pragma unroll
    for (int t = 0; t < 5; ++t) acc[t] = (v8f){0,0,0,0,0,0,0,0};

    for (int k0 = 0; k0 < D; k0 += 4) {
        int kA = k0 + half * 2;
        v2f a = *(const v2f*)(xrow + kA);                 // A[row, kA], A[row, kA+1]
        const float* w0 = W + (size_t)kA * LDH + n0 + l16;
        const float* w1 = w0 + LDH;
#pragma unroll
        for (int t = 0; t < 5; ++t) {
            v2f b;
            b.x = w0[t * 16];                             // B[kA,   col]
            b.y = w1[t * 16];                             // B[kA+1, col]
            acc[t] = __builtin_amdgcn_wmma_f32_16x16x4_f32(
                false, a, false, b, (short)0, acc[t], false, false);
        }
    }

    float* hbase = H + (size_t)m0 * LDH + n0 + l16;
#pragma unroll
    for (int v = 0; v < 8; ++v) {
        float* hr = hbase + (size_t)(v + half * 8) * LDH;
#pragma unroll
        for (int t = 0; t < 5; ++t) hr[t * 16] = acc[t][v];
    }
}

// ---------------- edge message + scatter-add ----------------
// One wave per edge. Lanes 0..24 each own a float4 chunk of D (25*4 = 100).
__global__ __launch_bounds__(256) void edge_scatter(const int* __restrict__ src,
                                                    const int* __restrict__ dst,
                                                    const int* __restrict__ etype,
                                                    const float* __restrict__ enorm,
                                                    const float* __restrict__ att,
                                                    const float* __restrict__ H,
                                                    float* __restrict__ aggr) {
    int e = (blockIdx.x * blockDim.x + threadIdx.x) >> 5;
    if (e >= E_EDGES) return;
    int lane = threadIdx.x & 31;
    if (lane >= 25) return;

    int   s = src[e];
    int   t = dst[e];
    int   r = etype[e];
    float nrm = enorm[e];

    const float* arow = att + (size_t)r * B_BASES;
    const float* hrow = H + (size_t)s * LDH + lane * 4;

    v4f m = (v4f){0, 0, 0, 0};
#pragma unroll
    for (int b = 0; b < B_BASES; ++b) {
        float c = arow[b] * nrm;
        v4f h = *(const v4f*)(hrow + b * 100);
        m.x += c * h.x; m.y += c * h.y; m.z += c * h.z; m.w += c * h.w;
    }
    float* out = aggr + (size_t)t * D + lane * 4;
    atomicAdd(out + 0, m.x);
    atomicAdd(out + 1, m.y);
    atomicAdd(out + 2, m.z);
    atomicAdd(out + 3, m.w);
}

// ---------------- finalize: mean + root + bias (+relu) ----------------
__global__ void finalize(const float* __restrict__ aggr, const float* __restrict__ deg,
                         const float* __restrict__ H, const float* __restrict__ bias,
                         float* __restrict__ out, int do_relu) {
    int i = blockIdx.x * blockDim.x + threadIdx.x;
    if (i >= NXD) return;
    int n = i / D, d = i - n * D;
    float dg = deg[n];
    dg = dg < 1.0f ? 1.0f : dg;
    float r = aggr[i] / dg + H[(size_t)n * LDH + 1000 + d] + bias[d];
    if (do_relu) r = fmaxf(r, 0.0f);
    out[i] = r;
}

// ---------------- host orchestration ----------------

extern "C" void kernel_launch(void* const* d_in, const int* in_sizes, int n_in,
                              void* d_out, int out_size, void* d_ws, size_t ws_size,
                              hipStream_t stream) {
    const int*   entity    = (const int*)d_in[0];
    const int*   edge_src  = (const int*)d_in[1];               // edge_index[0, :]
    const int*   edge_dst  = ((const int*)d_in[1]) + E_EDGES;   // edge_index[1, :]
    const int*   edge_type = (const int*)d_in[2];
    const float* edge_norm = (const float*)d_in[3];
    const float* emb       = (const float*)d_in[4];
    const float* basis[3]  = { (const float*)d_in[5],  (const float*)d_in[9],  (const float*)d_in[13] };
    const float* att[3]    = { (const float*)d_in[6],  (const float*)d_in[10], (const float*)d_in[14] };
    const float* root[3]   = { (const float*)d_in[7],  (const float*)d_in[11], (const float*)d_in[15] };
    const float* bias[3]   = { (const float*)d_in[8],  (const float*)d_in[12], (const float*)d_in[16] };
    float* out = (float*)d_out;

    // workspace layout (floats)
    float* ws   = (float*)d_ws;
    float* xA   = ws;                        // 5,000,000
    float* xB   = xA + NXD;                  // 5,000,000
    float* Hext = xB + NXD;                  // 50000*1120 = 56,000,000
    float* aggr = Hext + (size_t)N_NODES * LDH;  // 5,000,000
    float* deg  = aggr + NXD;                // 50,000
    float* Wp0  = deg + N_NODES;             // 3 x 112,000
    float* Wp1  = Wp0 + D * LDH;
    float* Wp2  = Wp1 + D * LDH;

    const int TB = 256;
    dim3 gNXD((NXD + TB - 1) / TB);
    dim3 gPack((D * LDH + TB - 1) / TB);
    dim3 gDeg((E_EDGES + TB - 1) / TB);
    dim3 gGemm((MTILES * NTILE_GROUPS * 32 + TB - 1) / TB);
    dim3 gEdge((E_EDGES * 32 + TB - 1) / TB);

    // setup: gather x0, pack weights, degree histogram (conv-invariant)
    gather_emb<<<gNXD, TB, 0, stream>>>(emb, entity, xA);
    pack_weights<<<gPack, TB, 0, stream>>>(basis[0], root[0], Wp0);
    pack_weights<<<gPack, TB, 0, stream>>>(basis[1], root[1], Wp1);
    pack_weights<<<gPack, TB, 0, stream>>>(basis[2], root[2], Wp2);
    zero_f32<<<(N_NODES + TB - 1) / TB, TB, 0, stream>>>(deg, N_NODES);
    degree_hist<<<gDeg, TB, 0, stream>>>(edge_dst, deg);

    // conv chain: layers [1,1,2,1,3] (0-indexed: 0,0,1,0,2), relu after conv #2 and #4
    struct Step { int layer; int relu; };
    const Step steps[5] = { {0,0}, {0,1}, {1,0}, {0,1}, {2,0} };
    const float* Wps[3] = { Wp0, Wp1, Wp2 };

    float* cur = xA;
    float* nxt = xB;
    for (int c = 0; c < 5; ++c) {
        int L = steps[c].layer;
        float* dstbuf = (c == 4) ? out : nxt;
        zero_f32<<<gNXD, TB, 0, stream>>>(aggr, NXD);
        gemm_wmma_f32<<<gGemm, TB, 0, stream>>>(cur, Wps[L], Hext);
        edge_scatter<<<gEdge, TB, 0, stream>>>(edge_src, edge_dst, edge_type, edge_norm,
                                               att[L], Hext, aggr);
        finalize<<<gNXD, TB, 0, stream>>>(aggr, deg, Hext, bias[L], dstbuf, steps[c].relu);
        float* tmp = cur; cur = dstbuf; nxt = tmp;
    }
}